// TransformerBlock_32564442038942
// MI455X (gfx1250) — compile-verified
//
#include <hip/hip_runtime.h>
#include <math.h>

typedef float v2f __attribute__((ext_vector_type(2)));
typedef float v8f __attribute__((ext_vector_type(8)));

// V_WMMA_F32_16X16X4_F32 : D = A(16x4) x B(4x16) + C(16x16), full fp32.
static __device__ __forceinline__ v8f wmma4(v2f a, v2f b, v8f c) {
  return __builtin_amdgcn_wmma_f32_16x16x4_f32(false, a, false, b, (short)0, c,
                                               false, false);
}

// Async global->LDS copy, 16B per lane. Tracked by ASYNCcnt.
static __device__ __forceinline__ void async_cp16(const float* g, float* l) {
  asm volatile("global_load_async_to_lds_b128 %0, %1, off"
               :
               : "v"((unsigned)(unsigned long long)l), "v"(g)
               : "memory");
}
static __device__ __forceinline__ void wait_async0() {
  asm volatile("s_wait_asynccnt 0x0" ::: "memory");
}

static __device__ __forceinline__ float redmax16(float v) {
  v = fmaxf(v, __shfl_xor(v, 1, 16));
  v = fmaxf(v, __shfl_xor(v, 2, 16));
  v = fmaxf(v, __shfl_xor(v, 4, 16));
  v = fmaxf(v, __shfl_xor(v, 8, 16));
  return v;
}
static __device__ __forceinline__ float redsum16(float v) {
  v += __shfl_xor(v, 1, 16);
  v += __shfl_xor(v, 2, 16);
  v += __shfl_xor(v, 4, 16);
  v += __shfl_xor(v, 8, 16);
  return v;
}

// GEMM K=256 with NT 16-wide column tiles. Wt is TRANSPOSED weights [N][256]:
// each B-operand fetch is one contiguous (aligned) b64 load; all NT tiles
// share one address register via immediate offsets (t*16KB + k*4).
template <int NT>
static __device__ __forceinline__ void gemm_tiles(const float* __restrict__ Arow,
                                                  const float* __restrict__ Wt,
                                                  int colbase, int n, int khalf,
                                                  v8f acc[NT]) {
  const float* wbase = Wt + (size_t)(colbase + n) * 256 + khalf;
  for (int k = 0; k < 256; k += 4) {
    const v2f a = *(const v2f*)(Arow + k + khalf);
#pragma unroll
    for (int t = 0; t < NT; ++t) {
      const v2f b = *(const v2f*)(wbase + t * 4096 + k);
      acc[t] = wmma4(a, b, acc[t]);
    }
  }
}

// ------------- Kernel 0: weight transpose W[K][N] -> Wt[N][K] ----------------
__global__ __launch_bounds__(256) void k_transpose(const float* __restrict__ W,
                                                   float* __restrict__ Wt,
                                                   int K, int N) {
  const int idx = blockIdx.x * 256 + threadIdx.x;
  if (idx < K * N) {
    const int k = idx / N;
    const int c = idx - k * N;
    Wt[(size_t)c * K + k] = W[idx];
  }
}

// ---------------- Kernel 1: QKV projection + scatter -------------------------
// Q,K stored [B*H, S, 128]; V stored TRANSPOSED [B*H, 128, S].
__global__ __launch_bounds__(256, 1) void k_qkv(const float* __restrict__ X,
                                                const float* __restrict__ WqkvT,
                                                const float* __restrict__ bias,
                                                float* __restrict__ Q,
                                                float* __restrict__ Kd,
                                                float* __restrict__ Vt) {
  const int wid = threadIdx.x >> 5;
  const int lane = threadIdx.x & 31;
  const int n = lane & 15;
  const int hi = lane >> 4;
  const int khalf = hi << 1;
  const int rowbase = blockIdx.x * 64 + (wid & 3) * 16;
  const int colbase = blockIdx.y * 256 + (wid >> 2) * 128;

  v8f acc[8] = {};
  const float* Arow = X + (size_t)(rowbase + n) * 256;
  gemm_tiles<8>(Arow, WqkvT, colbase, n, khalf, acc);

#pragma unroll
  for (int t = 0; t < 8; ++t) {
    const int col = colbase + t * 16 + n;   // 0..767
    const int h = col / 384;
    const int rem = col - h * 384;
    const int which = rem >> 7;             // 0=Q 1=K 2=V
    const int d = rem & 127;
    const float bb = bias[col];
#pragma unroll
    for (int r = 0; r < 8; ++r) {
      const int row = rowbase + r + (hi << 3);  // token index 0..16383
      const int b = row >> 11;
      const int s = row & 2047;
      const int bh = b * 2 + h;
      const float val = acc[t][r] + bb;
      if (which == 2) {
        Vt[((size_t)bh * 128 + d) * 2048 + s] = val;
      } else {
        float* dst = (which == 0) ? Q : Kd;
        dst[((size_t)bh * 2048 + s) * 128 + d] = val;
      }
    }
  }
}

// ---------------- Kernel 2: flash attention with async double-buffering ------
// Workgroup cooperatively stages each 16-key K tile and V tile into LDS via
// GLOBAL_LOAD_ASYNC_TO_LDS_B128 (double-buffered, ASYNCcnt), then all 8 waves
// consume them with conflict-free ds_load_b64 (K rows stride 132f, V rows
// stride 20f -> lanes tile the 64 LDS banks disjointly).
__global__ __launch_bounds__(256, 1) void k_attn(const float* __restrict__ Q,
                                                 const float* __restrict__ Kd,
                                                 const float* __restrict__ Vt,
                                                 float* __restrict__ O) {
  const int tid = threadIdx.x;
  const int wid = tid >> 5;
  const int lane = tid & 31;
  const int n = lane & 15;
  const int hi = lane >> 4;
  const int khalf = hi << 1;
  const int bh = blockIdx.y;        // b*2 + h
  const int h = bh & 1;
  const int b = bh >> 1;
  const int qrow0 = blockIdx.x * 128 + wid * 16;

  const float* Qp = Q + (size_t)bh * 2048 * 128;
  const float* Kp = Kd + (size_t)bh * 2048 * 128;
  const float* Vp = Vt + (size_t)bh * 128 * 2048;  // [128][2048]

  // LDS: K double buffer (16 x stride132), V double buffer (128 x stride20),
  // per-wave P staging (16 x stride17).
  __shared__ __align__(16) float smem[11520];
  float* kbuf0 = smem;                 // 2112 floats
  float* kbuf1 = smem + 2112;
  float* vbuf0 = smem + 4224;          // 2560 floats
  float* vbuf1 = smem + 6784;
  float* pl = smem + 9344 + wid * 272;

  // This wave's 16x128 Q tile in A-operand registers (32 k-steps of K=4)
  v2f aq[32];
  const float* qr = Qp + (size_t)(qrow0 + n) * 128 + khalf;
#pragma unroll
  for (int kk = 0; kk < 32; ++kk) aq[kk] = *(const v2f*)(qr + 4 * kk);

  v8f acc[8] = {};                    // 16 rows x 128 cols output accumulator
  float mrow[8], lrow[8];
#pragma unroll
  for (int r = 0; r < 8; ++r) { mrow[r] = -__builtin_inff(); lrow[r] = 0.f; }

  // Cooperative stage of key tile j into buffer c: 1024 lane-transfers of 16B
  // (K: 16 rows x 32 chunks, V: 128 rows x 4 chunks) = 4 asyncs per thread.
  auto stage = [&](int j, int c) {
    float* kb = c ? kbuf1 : kbuf0;
    float* vb = c ? vbuf1 : vbuf0;
#pragma unroll
    for (int i = 0; i < 4; ++i) {
      const int idx = i * 256 + tid;
      if (idx < 512) {
        const int row = idx >> 5, chunk = idx & 31;
        async_cp16(Kp + (size_t)(j * 16 + row) * 128 + chunk * 4,
                   kb + row * 132 + chunk * 4);
      } else {
        const int v = idx - 512;
        const int row = v >> 2, chunk = v & 3;
        async_cp16(Vp + (size_t)row * 2048 + j * 16 + chunk * 4,
                   vb + row * 20 + chunk * 4);
      }
    }
  };

  stage(0, 0);
  wait_async0();
  __syncthreads();

  for (int j = 0; j < 128; ++j) {
    const int c = j & 1;
    if (j + 1 < 128) stage(j + 1, c ^ 1);  // prefetch overlaps tile compute

    // scores S = Q x K^T (16x16), K=128 via 32 WMMAs; B operand from LDS
    v8f sc = {};
    const float* kb = (c ? kbuf1 : kbuf0) + n * 132 + khalf;
#pragma unroll
    for (int kk = 0; kk < 32; ++kk) {
      const v2f bk = *(const v2f*)(kb + 4 * kk);
      sc = wmma4(aq[kk], bk, sc);
    }
    // online softmax; row r lives in half-wave group
#pragma unroll
    for (int r = 0; r < 8; ++r) {
      const float v = sc[r];
      const float rmax = redmax16(v);
      const float mnew = fmaxf(mrow[r], rmax);
      const float scale = __expf(mrow[r] - mnew);
      const float p = __expf(v - mnew);
      lrow[r] = lrow[r] * scale + redsum16(p);
      mrow[r] = mnew;
#pragma unroll
      for (int t = 0; t < 8; ++t) acc[t][r] *= scale;
      pl[(r + hi * 8) * 17 + n] = p;  // stage P for transpose to A-layout
    }
    asm volatile("s_wait_dscnt 0x0" ::: "memory");
    v2f pa[4];
#pragma unroll
    for (int kk = 0; kk < 4; ++kk) {
      pa[kk].x = pl[n * 17 + 4 * kk + khalf];
      pa[kk].y = pl[n * 17 + 4 * kk + khalf + 1];
    }
    asm volatile("s_wait_dscnt 0x0" ::: "memory");
    // acc += P x V ; V tile in LDS as [d][16 keys], stride 20
    const float* vb = (c ? vbuf1 : vbuf0) + n * 20 + khalf;
#pragma unroll
    for (int t = 0; t < 8; ++t) {
#pragma unroll
      for (int kk = 0; kk < 4; ++kk) {
        const v2f bv = *(const v2f*)(vb + t * 320 + 4 * kk);
        acc[t] = wmma4(pa[kk], bv, acc[t]);
      }
    }
    wait_async0();    // prefetch landed in LDS
    __syncthreads();  // all waves done with buffer c; flip
  }
  // normalize and store into attn_out [B,S,256] (col = h*128 + d)
#pragma unroll
  for (int t = 0; t < 8; ++t) {
#pragma unroll
    for (int r = 0; r < 8; ++r) {
      const int row = qrow0 + r + hi * 8;
      O[((size_t)(b * 2048 + row)) * 256 + h * 128 + t * 16 + n] =
          acc[t][r] / lrow[r];
    }
  }
}

// ------------- Kernel 3/5: GEMM + bias + ReLU + residual + LayerNorm ---------
__global__ __launch_bounds__(256, 1) void k_proj_ln(
    const float* __restrict__ A, const float* __restrict__ Wt,
    const float* __restrict__ bias, const float* __restrict__ residual,
    const float* __restrict__ gamma, const float* __restrict__ beta,
    float* __restrict__ out) {
  const int wid = threadIdx.x >> 5;
  const int lane = threadIdx.x & 31;
  const int n = lane & 15;
  const int hi = lane >> 4;
  const int khalf = hi << 1;
  const int rowbase = blockIdx.x * 128 + wid * 16;

  v8f acc[16] = {};
  const float* Arow = A + (size_t)(rowbase + n) * 256;
  gemm_tiles<16>(Arow, Wt, 0, n, khalf, acc);

  float sum[8] = {}, ssq[8] = {};
#pragma unroll
  for (int t = 0; t < 16; ++t) {
    const int col = t * 16 + n;
    const float bb = bias[col];
#pragma unroll
    for (int r = 0; r < 8; ++r) {
      const int row = rowbase + r + hi * 8;
      float v = fmaxf(acc[t][r] + bb, 0.f);            // ReLU
      v += residual[(size_t)row * 256 + col];          // residual
      acc[t][r] = v;
      sum[r] += v;
      ssq[r] += v * v;
    }
  }
  float mean[8], rstd[8];
#pragma unroll
  for (int r = 0; r < 8; ++r) {
    const float s = redsum16(sum[r]);
    const float q = redsum16(ssq[r]);
    mean[r] = s * (1.f / 256.f);
    const float var = q * (1.f / 256.f) - mean[r] * mean[r];
    rstd[r] = __frsqrt_rn(var + 1e-5f);
  }
#pragma unroll
  for (int t = 0; t < 16; ++t) {
    const int col = t * 16 + n;
    const float g = gamma[col];
    const float be = beta[col];
#pragma unroll
    for (int r = 0; r < 8; ++r) {
      const int row = rowbase + r + hi * 8;
      out[(size_t)row * 256 + col] = (acc[t][r] - mean[r]) * rstd[r] * g + be;
    }
  }
}

// ---------------- Kernel 4: GEMM + bias + ReLU (FF1) -------------------------
__global__ __launch_bounds__(256, 1) void k_ff1(const float* __restrict__ A,
                                                const float* __restrict__ Wt,
                                                const float* __restrict__ bias,
                                                float* __restrict__ out) {
  const int wid = threadIdx.x >> 5;
  const int lane = threadIdx.x & 31;
  const int n = lane & 15;
  const int hi = lane >> 4;
  const int khalf = hi << 1;
  const int rowbase = blockIdx.x * 128 + wid * 16;

  v8f acc[16] = {};
  const float* Arow = A + (size_t)(rowbase + n) * 256;
  gemm_tiles<16>(Arow, Wt, 0, n, khalf, acc);

#pragma unroll
  for (int t = 0; t < 16; ++t) {
    const int col = t * 16 + n;
    const float bb = bias[col];
#pragma unroll
    for (int r = 0; r < 8; ++r) {
      const int row = rowbase + r + hi * 8;
      out[(size_t)row * 256 + col] = fmaxf(acc[t][r] + bb, 0.f);
    }
  }
}

extern "C" void kernel_launch(void* const* d_in, const int* in_sizes, int n_in,
                              void* d_out, int out_size, void* d_ws,
                              size_t ws_size, hipStream_t stream) {
  const float* x = (const float*)d_in[0];
  const float* Wqkv = (const float*)d_in[1];
  const float* bqkv = (const float*)d_in[2];
  const float* Wattn = (const float*)d_in[3];
  const float* battn = (const float*)d_in[4];
  const float* W1 = (const float*)d_in[5];
  const float* b1 = (const float*)d_in[6];
  const float* W2 = (const float*)d_in[7];
  const float* b2 = (const float*)d_in[8];
  const float* g1 = (const float*)d_in[9];
  const float* be1 = (const float*)d_in[10];
  const float* g2 = (const float*)d_in[11];
  const float* be2 = (const float*)d_in[12];

  const size_t NTOK = 8 * 2048;          // 16384 tokens
  const size_t SEG = NTOK * 256;         // 4,194,304 floats
  float* ws = (float*)d_ws;
  float* Q = ws;
  float* K = ws + SEG;
  float* Vt = ws + 2 * SEG;              // transposed V [B*H,128,S]
  float* attnO = ws + 3 * SEG;
  float* x1 = ws + 4 * SEG;
  float* hid = ws + 5 * SEG;
  float* WqkvT = ws + 6 * SEG;           // 768*256
  float* WattnT = WqkvT + 768 * 256;     // 256*256
  float* W1T = WattnT + 256 * 256;
  float* W2T = W1T + 256 * 256;
  float* out = (float*)d_out;

  // 0) transpose weights to [N][K] so B-operands are contiguous b64 loads
  k_transpose<<<dim3(768), 256, 0, stream>>>(Wqkv, WqkvT, 256, 768);
  k_transpose<<<dim3(256), 256, 0, stream>>>(Wattn, WattnT, 256, 256);
  k_transpose<<<dim3(256), 256, 0, stream>>>(W1, W1T, 256, 256);
  k_transpose<<<dim3(256), 256, 0, stream>>>(W2, W2T, 256, 256);

  // 1) QKV projection: [16384,256] x [256,768]
  k_qkv<<<dim3(256, 3), 256, 0, stream>>>(x, WqkvT, bqkv, Q, K, Vt);
  // 2) flash attention per (b,h), 128 query rows per WG, async K/V staging
  k_attn<<<dim3(16, 16), 256, 0, stream>>>(Q, K, Vt, attnO);
  // 3) attn proj + ReLU + residual(x) + LN1 -> x1
  k_proj_ln<<<dim3(128), 256, 0, stream>>>(attnO, WattnT, battn, x, g1, be1, x1);
  // 4) FF1 + ReLU -> hid
  k_ff1<<<dim3(128), 256, 0, stream>>>(x1, W1T, b1, hid);
  // 5) FF2 + ReLU + residual(x1) + LN2 -> out
  k_proj_ln<<<dim3(128), 256, 0, stream>>>(hid, W2T, b2, x1, g2, be2, out);
}